// a_std_op_6631429505492
// MI455X (gfx1250) — compile-verified
//
#include <hip/hip_runtime.h>
#include <hip/hip_bf16.h>
#include <stdint.h>

// ---------------------------------------------------------------------------
// Per-node std over incoming-edge features (segment sum / sumsq / count).
//   d_in[0]: src_emb    [nE, 32] f32
//   d_in[1]: src_emb_in [nE, 32] f32  (unused by reference)
//   d_in[2]: dst_idx    [nE]     int
//   d_in[3]: num_nodes  scalar   int
//   d_out  : [N, 32] f32
// Workspace layout (floats): sum[N*32] | sumsq[N*32] | cnt[N]
// ---------------------------------------------------------------------------

#define D_FEAT 32
#define TILE   256   // edges per block tile; 256*32*4 = 32 KB LDS

typedef __attribute__((ext_vector_type(4))) unsigned int v4u;
typedef __attribute__((ext_vector_type(8))) int          v8i;
typedef __attribute__((ext_vector_type(4))) int          v4i;

__global__ void nodestd_zero_ws(float* __restrict__ ws, size_t n) {
    size_t i = (size_t)blockIdx.x * blockDim.x + threadIdx.x;
    size_t stride = (size_t)gridDim.x * blockDim.x;
    for (; i < n; i += stride) ws[i] = 0.0f;
}

__global__ __launch_bounds__(256)
void nodestd_accum(const float* __restrict__ src,
                   const int*   __restrict__ dst,
                   float* __restrict__ sum,
                   float* __restrict__ sq,
                   float* __restrict__ cnt,
                   int nE) {
    __shared__ float tile[TILE * D_FEAT];

    const int tid  = (int)threadIdx.x;
    const int e0   = (int)blockIdx.x * TILE;

    // Warm the index lines for this tile while the TDM fills LDS.
    if (e0 + tid < nE) __builtin_prefetch(&dst[e0 + tid], 0, 0);

    // Wave 0 drives the Tensor Data Mover: DMA a [rows x 32] f32 tile of
    // src_emb straight into LDS. All descriptor fields are block-uniform
    // (blockIdx + kernel args only) so they live in SGPRs.
    if (tid < 32) {
        const uint64_t gaddr = (uint64_t)(uintptr_t)(src + (size_t)e0 * D_FEAT);
        const uint32_t ldsAddr = (uint32_t)(uintptr_t)(&tile[0]);
        const uint32_t rem = (uint32_t)(nE - e0);   // rows left in tensor from tile start

        // D# group 0: count=1 | lds_addr | global_addr[56:0] | type=2
        v4u g0;
        g0.x = 1u;                                   // count=1 (valid), user mode
        g0.y = ldsAddr;                              // LDS byte address
        g0.z = (uint32_t)(gaddr & 0xFFFFFFFFu);      // global_addr[31:0]
        g0.w = (uint32_t)((gaddr >> 32) & 0x1FFFFFFu) | (2u << 30); // [56:32] | type=2

        // D# group 1: data_size=4B, tensor_dim0=32, tensor_dim1=rem,
        //             tile_dim0=32, tile_dim1=TILE, dim0_stride=32 elements.
        v8i g1;
        g1[0] = (int)(2u << 16);                     // workgroup_mask=0, data_size=2 (4B)
        g1[1] = (int)((uint32_t)D_FEAT << 16);       // tensor_dim0[15:0]=32
        g1[2] = (int)((rem & 0xFFFFu) << 16);        // tensor_dim0[31:16]=0 | tensor_dim1[15:0]
        g1[3] = (int)(((rem >> 16) & 0xFFFFu) |      // tensor_dim1[31:16]
                      ((uint32_t)D_FEAT << 16));     // tile_dim0=32
        g1[4] = TILE;                                // tile_dim1=TILE, tile_dim2=0
        g1[5] = D_FEAT;                              // tensor_dim0_stride[31:0]=32
        g1[6] = 0;                                   // stride hi | tensor_dim1_stride lo (unused, 2D)
        g1[7] = 0;

        v4i g2 = {0, 0, 0, 0};                       // 2-D tensor: groups 2/3 unused
        v4i g3 = {0, 0, 0, 0};
        v8i g4 = {0, 0, 0, 0, 0, 0, 0, 0};           // extra group (clang-23 6-arg form)

        __builtin_amdgcn_tensor_load_to_lds(g0, g1, g2, g3, g4, 0);
        __builtin_amdgcn_s_wait_tensorcnt(0);
    }
    __syncthreads();

    // 8 waves x 32 lanes: lane = feature, wave strides over tile rows.
    const int lane = tid & 31;
    const int wave = tid >> 5;
    for (int r = wave; r < TILE; r += 8) {
        const int e = e0 + r;
        if (e >= nE) break;
        const int d = dst[e];                        // lane-uniform broadcast load
        const float v = tile[r * D_FEAT + lane];
        const size_t o = (size_t)d * D_FEAT + lane;
        atomicAdd(&sum[o], v);
        atomicAdd(&sq[o], v * v);
        if (lane == 0) atomicAdd(&cnt[d], 1.0f);
    }
}

__global__ void nodestd_finalize(const float* __restrict__ sum,
                                 const float* __restrict__ sq,
                                 const float* __restrict__ cnt,
                                 float* __restrict__ out,
                                 int total) {
    int i = (int)(blockIdx.x * blockDim.x + threadIdx.x);
    if (i >= total) return;
    const int n = i >> 5;                            // node
    const float c = cnt[n];
    const float denom = fmaxf(c, 1.0f);
    const float mean  = sum[i] / denom;
    const float msq   = sq[i] / denom;
    const float var   = fmaxf(msq - mean * mean, 0.0f);
    out[i] = (c > 0.0f) ? sqrtf(var + 1e-5f) : 0.0f;
}

extern "C" void kernel_launch(void* const* d_in, const int* in_sizes, int n_in,
                              void* d_out, int out_size, void* d_ws, size_t ws_size,
                              hipStream_t stream) {
    const float* src = (const float*)d_in[0];
    const int*   dst = (const int*)d_in[2];
    float* out = (float*)d_out;

    const int nE = in_sizes[0] / D_FEAT;     // 1,600,000
    const int N  = out_size / D_FEAT;        // 100,000

    float* ws_sum = (float*)d_ws;
    float* ws_sq  = ws_sum + (size_t)N * D_FEAT;
    float* ws_cnt = ws_sq  + (size_t)N * D_FEAT;
    const size_t ws_elems = (size_t)N * (2 * D_FEAT + 1);

    // Pass 0: zero sums/sumsq/count.
    {
        int blocks = (int)((ws_elems + 255) / 256);
        if (blocks > 4096) blocks = 4096;
        nodestd_zero_ws<<<blocks, 256, 0, stream>>>((float*)d_ws, ws_elems);
    }

    // Pass 1: TDM-staged scatter accumulate.
    {
        int blocks = (nE + TILE - 1) / TILE;  // 6250 for 1.6M edges
        nodestd_accum<<<blocks, 256, 0, stream>>>(src, dst, ws_sum, ws_sq, ws_cnt, nE);
    }

    // Pass 2: finalize std per (node, feature).
    {
        int total = N * D_FEAT;
        int blocks = (total + 255) / 256;
        nodestd_finalize<<<blocks, 256, 0, stream>>>(ws_sum, ws_sq, ws_cnt, out, total);
    }
}